// GraphNeuralNetwork_64647847739561
// MI455X (gfx1250) — compile-verified
//
#include <hip/hip_runtime.h>
#include <hip/hip_bf16.h>

typedef __attribute__((ext_vector_type(2))) float v2f;
typedef __attribute__((ext_vector_type(8))) float v8f;

// ---------------------------------------------------------------------------
// Kernel 1: zero the workspace (agg_sum 50000x128 + deg 50000)
// ---------------------------------------------------------------------------
__global__ void gnn_zero_kernel(float* __restrict__ p, long long n) {
    long long i = (long long)blockIdx.x * blockDim.x + threadIdx.x;
    long long stride = (long long)gridDim.x * blockDim.x;
    for (; i < n; i += stride) p[i] = 0.0f;
}

// ---------------------------------------------------------------------------
// Kernel 2: symmetric edge scatter.
// One wave (32 lanes) per edge. Each lane owns a float4 slice of the 128-wide
// feature row -> coalesced 512B loads and coalesced hardware f32 atomics.
// agg[s] += x[d]; agg[d] += x[s]; deg[s] += 1; deg[d] += 1.
// ---------------------------------------------------------------------------
__device__ __forceinline__ void atomAddF(float* p, float v) {
    unsafeAtomicAdd(p, v);   // hardware global_atomic_add_f32 (agg/deg are device-local)
}

__global__ void gnn_scatter_kernel(const float* __restrict__ x,
                                   const long long* __restrict__ edges,
                                   float* __restrict__ agg,
                                   float* __restrict__ deg,
                                   int n_edges) {
    int wave = (int)((blockIdx.x * (long long)blockDim.x + threadIdx.x) >> 5);
    int lane = threadIdx.x & 31;
    if (wave >= n_edges) return;

    long long s = edges[2LL * wave + 0];
    long long d = edges[2LL * wave + 1];

    const float4 xs = *(const float4*)(x + s * 128 + lane * 4);
    const float4 xd = *(const float4*)(x + d * 128 + lane * 4);

    float* as = agg + s * 128 + lane * 4;
    float* ad = agg + d * 128 + lane * 4;

    atomAddF(as + 0, xd.x); atomAddF(as + 1, xd.y);
    atomAddF(as + 2, xd.z); atomAddF(as + 3, xd.w);
    atomAddF(ad + 0, xs.x); atomAddF(ad + 1, xs.y);
    atomAddF(ad + 2, xs.z); atomAddF(ad + 3, xs.w);

    if (lane == 0) {
        atomAddF(deg + s, 1.0f);
        atomAddF(deg + d, 1.0f);
    }
}

// ---------------------------------------------------------------------------
// Kernel 3: fused concat-GEMM with fp32 WMMA.
//   out[16-row strip] = [agg/deg | x] (16x256) @ [W_neigh; W_self] (256x128) + b
// Block = 256 threads (8 waves). Wave w computes columns [16w,16w+16).
// A strip staged in LDS (pad stride 260 -> conflict-free b64 reads).
// WMMA f32 16x16x4 fragment layouts (ISA 7.12.2):
//   A (16x4):  lane L holds row M=L%16, K = {2*(L/16), 2*(L/16)+1}
//   B (4x16):  lane L holds col N=L%16, K = {2*(L/16), 2*(L/16)+1}
//   C/D:       VGPR v, lane L -> M = v + 8*(L/16), N = L%16
// ---------------------------------------------------------------------------
__global__ __launch_bounds__(256)
void gnn_gemm_kernel(const float* __restrict__ agg,
                     const float* __restrict__ deg,
                     const float* __restrict__ x,
                     const float* __restrict__ Wn,
                     const float* __restrict__ Ws,
                     const float* __restrict__ bias,
                     float* __restrict__ out,
                     int n_nodes) {
    __shared__ float As[16][260];   // 16 rows x 256 K, +4 pad floats
    __shared__ float sInv[16];

    const int tid  = threadIdx.x;
    const int row0 = blockIdx.x * 16;

    if (tid < 16) {
        int r = row0 + tid;
        float dg = (r < n_nodes) ? deg[r] : 1.0f;
        sInv[tid] = 1.0f / fmaxf(dg, 1.0f);
    }
    __syncthreads();

    // Stage A = [agg_mean | x]; iteration j loads full row j with k = tid.
    #pragma unroll 4
    for (int j = 0; j < 16; ++j) {
        int r  = row0 + j;
        int rc = (r < n_nodes) ? r : (n_nodes - 1);
        int k  = tid;
        float v;
        if (k < 128) v = agg[(long long)rc * 128 + k] * sInv[j];
        else         v = x[(long long)rc * 128 + (k - 128)];
        As[j][k] = v;
    }
    __syncthreads();

    const int wave  = tid >> 5;            // 0..7 -> column tile
    const int lane  = tid & 31;
    const int half  = lane >> 4;           // 0 or 1
    const int rowA  = lane & 15;
    const int col   = wave * 16 + (lane & 15);
    const int khalf = half * 2;            // K offset within the 4-wide step

    v8f c = {0.0f, 0.0f, 0.0f, 0.0f, 0.0f, 0.0f, 0.0f, 0.0f};

    // K = 0..127 : W_neigh  (A from As[:, 0..127])
    #pragma unroll 8
    for (int s = 0; s < 32; ++s) {
        const int ks = 4 * s + khalf;
        v2f a, b;
        a[0] = As[rowA][ks];
        a[1] = As[rowA][ks + 1];
        b[0] = Wn[ks * 128 + col];
        b[1] = Wn[(ks + 1) * 128 + col];
        c = __builtin_amdgcn_wmma_f32_16x16x4_f32(false, a, false, b,
                                                  (short)0, c, false, false);
    }
    // K = 128..255 : W_self  (A from As[:, 128..255])
    #pragma unroll 8
    for (int s = 0; s < 32; ++s) {
        const int ks = 4 * s + khalf;
        v2f a, b;
        a[0] = As[rowA][128 + ks];
        a[1] = As[rowA][128 + ks + 1];
        b[0] = Ws[ks * 128 + col];
        b[1] = Ws[(ks + 1) * 128 + col];
        c = __builtin_amdgcn_wmma_f32_16x16x4_f32(false, a, false, b,
                                                  (short)0, c, false, false);
    }

    const float bv = bias[col];
    #pragma unroll
    for (int v = 0; v < 8; ++v) {
        int r = row0 + v + half * 8;
        if (r < n_nodes) out[(long long)r * 128 + col] = c[v] + bv;
    }
}

// ---------------------------------------------------------------------------
extern "C" void kernel_launch(void* const* d_in, const int* in_sizes, int n_in,
                              void* d_out, int out_size, void* d_ws, size_t ws_size,
                              hipStream_t stream) {
    const float*      x     = (const float*)d_in[0];
    const long long*  edges = (const long long*)d_in[1];   // int64 edge_index
    const float*      Wn    = (const float*)d_in[2];
    const float*      Ws    = (const float*)d_in[3];
    const float*      bias  = (const float*)d_in[4];
    float*            out   = (float*)d_out;

    const int n_nodes = in_sizes[0] / 128;
    const int n_edges = in_sizes[1] / 2;

    float* agg = (float*)d_ws;                       // n_nodes*128 floats
    float* deg = agg + (size_t)n_nodes * 128;        // n_nodes floats

    // 1) zero workspace
    long long n_zero = (long long)n_nodes * 128 + n_nodes;
    int zgrid = (int)((n_zero + 255) / 256);
    if (zgrid > 16384) zgrid = 16384;
    gnn_zero_kernel<<<zgrid, 256, 0, stream>>>(agg, n_zero);

    // 2) edge scatter: 8 edges (waves) per 256-thread block
    int sgrid = (n_edges + 7) / 8;
    gnn_scatter_kernel<<<sgrid, 256, 0, stream>>>(x, edges, agg, deg, n_edges);

    // 3) fused concat-GEMM with fp32 WMMA: one block per 16-row strip
    int ggrid = (n_nodes + 15) / 16;
    gnn_gemm_kernel<<<ggrid, 256, 0, stream>>>(agg, deg, x, Wn, Ws, bias, out, n_nodes);
}